// GCNPolicy_27084063768597
// MI455X (gfx1250) — compile-verified
//
#include <hip/hip_runtime.h>

typedef __attribute__((ext_vector_type(2))) float v2f;
typedef __attribute__((ext_vector_type(8))) float v8f;

#define N_NODES   131072
#define E_EDGES   4194304
#define S_IN      32
#define H_DIM     16
#define B_GRAPHS  128
#define FC_IN_SZ  16384
#define KCHUNKS   8            // split-K factor for the big FC GEMMs

// ---------------------------------------------------------------- utilities
__global__ void fill_kernel(float* __restrict__ p, float v, int n) {
    int i = blockIdx.x * blockDim.x + threadIdx.x;
    if (i < n) p[i] = v;
}

// deg[c] += 1 for every edge target (deg pre-filled with 1.0 for self-loop)
__global__ void degree_kernel(const int* __restrict__ col, float* __restrict__ deg, int e) {
    int i = blockIdx.x * blockDim.x + threadIdx.x;
    if (i < e) atomicAdd(&deg[col[i]], 1.0f);
}

// in-place deg -> 1/sqrt(deg)   (deg >= 1 always, so no zero guard needed)
__global__ void rsqrt_kernel(float* __restrict__ d, int n) {
    int i = blockIdx.x * blockDim.x + threadIdx.x;
    if (i < n) d[i] = rsqrtf(d[i]);
}

// ---------------------------------------------------------------- WMMA GEMM
// C[M,N] = A[M,K] @ B[K,N], f32 via V_WMMA_F32_16X16X4_F32.
// One wave per 16x16 output tile per K-chunk. grid = (N/16, M/16, chunks), block = 32.
// chunks == 1 : direct store with bias (+optional relu).
// chunks  > 1 : each chunk z covers K-range [z*Kc, (z+1)*Kc) and stores a raw
//               partial at C + z*M*N; epilogue kernel reduces deterministically.
// A-frag: lane<16 -> (M=lane,   K=k+0..1), lane>=16 -> (M=lane-16, K=k+2..3)
// B-frag: lane<16 -> (K=k+v,    N=lane),   lane>=16 -> (K=k+2+v,   N=lane-16)
// C/D   : VGPR v, lanes 0-15 -> M=v, lanes 16-31 -> M=v+8
__global__ __launch_bounds__(32)
void wmma_gemm_f32(const float* __restrict__ A, const float* __restrict__ B,
                   const float* __restrict__ bias, float* __restrict__ C,
                   int K, int Nld, int relu, int Kc) {
    const int nb   = blockIdx.x;
    const int mb   = blockIdx.y;
    const int z    = blockIdx.z;
    const int lane = threadIdx.x;
    const int half = lane >> 4;     // 0 or 1
    const int lm   = lane & 15;

    const int k0 = z * Kc;

    v8f acc = {};
    const float* Ap = A + (size_t)(mb * 16 + lm) * K + k0 + 2 * half;   // 8B-aligned
    const float* Bp = B + (size_t)(k0 + 2 * half) * Nld + nb * 16 + lm;

#pragma unroll 4
    for (int k = 0; k < Kc; k += 4) {
        v2f a = *(const v2f*)Ap;            // K=k+2h, k+2h+1 for this lane's row
        v2f b;
        b.x = Bp[0];                        // K=k+2h   row of B
        b.y = Bp[Nld];                      // K=k+2h+1 row of B
        acc = __builtin_amdgcn_wmma_f32_16x16x4_f32(
                  false, a, false, b, (short)0, acc, false, false);
        Ap += 4;
        Bp += (size_t)4 * Nld;
    }

    const int cn = nb * 16 + lm;
    if (gridDim.z == 1) {
        const float bb = bias ? bias[cn] : 0.0f;
#pragma unroll
        for (int v = 0; v < 8; ++v) {
            float r = acc[v] + bb;
            if (relu) r = fmaxf(r, 0.0f);
            C[(size_t)(mb * 16 + v + 8 * half) * Nld + cn] = r;
        }
    } else {
        const size_t M  = (size_t)gridDim.y * 16;
        float* Cp = C + (size_t)z * M * Nld;
#pragma unroll
        for (int v = 0; v < 8; ++v) {
            Cp[(size_t)(mb * 16 + v + 8 * half) * Nld + cn] = acc[v];
        }
    }
}

// deterministic split-K reduction + bias + relu : C[i] = relu(sum_z part[z][i] + bias[i%N])
__global__ void reduce_bias_relu_kernel(const float* __restrict__ part,
                                        const float* __restrict__ bias,
                                        float* __restrict__ C,
                                        int MN, int Nld, int relu) {
    int i = blockIdx.x * blockDim.x + threadIdx.x;
    if (i >= MN) return;
    float s = 0.0f;
#pragma unroll
    for (int zc = 0; zc < KCHUNKS; ++zc) s += part[(size_t)zc * MN + i];
    s += bias[i % Nld];
    if (relu) s = fmaxf(s, 0.0f);
    C[i] = s;
}

// ---------------------------------------------------------------- GCN scatter
// out[col[e], :] += dinv[row]*dinv[col] * xw[row[e], :] ; 16 lanes per edge
__global__ void scatter_kernel(const float* __restrict__ xw,
                               const int* __restrict__ row, const int* __restrict__ col,
                               const float* __restrict__ dinv,
                               float* __restrict__ out, int e16) {
    int t = blockIdx.x * blockDim.x + threadIdx.x;
    if (t >= e16) return;
    int e = t >> 4, j = t & 15;
    int r = row[e], c = col[e];
    float w = dinv[r] * dinv[c];
    atomicAdd(&out[c * H_DIM + j], w * xw[r * H_DIM + j]);
}

// h = relu(scatter + dinv^2 * xw_self + bias)
__global__ void finalize_kernel(float* __restrict__ h, const float* __restrict__ xw,
                                const float* __restrict__ dinv,
                                const float* __restrict__ bias, int n) {
    int i = blockIdx.x * blockDim.x + threadIdx.x;
    if (i >= n) return;
    int node = i >> 4, j = i & 15;
    float d = dinv[node];
    float r = h[i] + d * d * xw[i] + bias[j];
    h[i] = fmaxf(r, 0.0f);
}

// ---------------------------------------------------------------- value heads
__global__ void heads_kernel(const float* __restrict__ V,
                             const float* __restrict__ Wiv, const float* __restrict__ biv,
                             const float* __restrict__ Wev, const float* __restrict__ bev,
                             float* __restrict__ out_ext, float* __restrict__ out_int) {
    int r = blockIdx.x * blockDim.x + threadIdx.x;
    if (r >= B_GRAPHS) return;
    const float* v = V + (size_t)r * 256;
    float si = 0.0f, se = 0.0f;
#pragma unroll 4
    for (int k = 0; k < 256; ++k) {
        float x = v[k];
        si += x * Wiv[k];
        se += x * Wev[k];
    }
    out_int[r] = si + biv[0];
    out_ext[r] = se + bev[0];
}

// ---------------------------------------------------------------- launcher
extern "C" void kernel_launch(void* const* d_in, const int* in_sizes, int n_in,
                              void* d_out, int out_size, void* d_ws, size_t ws_size,
                              hipStream_t stream) {
    const float* x    = (const float*)d_in[0];
    const int*   ei   = (const int*)d_in[1];        // [2, E] int32
    const float* W1   = (const float*)d_in[2];
    const float* b1   = (const float*)d_in[3];
    const float* W2   = (const float*)d_in[4];
    const float* b2   = (const float*)d_in[5];
    const float* W3   = (const float*)d_in[6];
    const float* b3   = (const float*)d_in[7];
    const float* Wp1  = (const float*)d_in[8];
    const float* bp1  = (const float*)d_in[9];
    const float* Wp2  = (const float*)d_in[10];
    const float* bp2  = (const float*)d_in[11];
    const float* Wv1  = (const float*)d_in[12];
    const float* bv1  = (const float*)d_in[13];
    const float* Wiv  = (const float*)d_in[14];
    const float* biv  = (const float*)d_in[15];
    const float* Wev  = (const float*)d_in[16];
    const float* bev  = (const float*)d_in[17];

    const int* row = ei;              // edge_index[0]
    const int* col = ei + E_EDGES;    // edge_index[1]

    // workspace layout
    char* ws = (char*)d_ws;
    float* dinv = (float*)ws;                                  // N floats
    float* bufA = (float*)(ws + (size_t)N_NODES * 4);          // N*16 floats (xw)
    float* bufB = bufA + (size_t)N_NODES * H_DIM;              // N*16 floats (h)
    float* Hp   = bufB + (size_t)N_NODES * H_DIM;              // 128*256
    float* Vv   = Hp + (size_t)B_GRAPHS * 256;                 // 128*256
    float* part = Vv + (size_t)B_GRAPHS * 256;                 // KCHUNKS*128*256

    float* P     = (float*)d_out;                 // [128,1024]
    float* ext_V = P + (size_t)B_GRAPHS * 1024;   // [128]
    float* int_V = ext_V + B_GRAPHS;              // [128]

    const int TPB = 256;
    const int nh  = N_NODES * H_DIM;              // 2M elements
    const int e16 = E_EDGES * H_DIM;              // 67M lane-tasks
    const int fcMN = B_GRAPHS * 256;              // 32768

    // --- symmetric degree norm (shared by all 3 convs) ---
    fill_kernel<<<(N_NODES + TPB - 1) / TPB, TPB, 0, stream>>>(dinv, 1.0f, N_NODES);
    degree_kernel<<<(E_EDGES + TPB - 1) / TPB, TPB, 0, stream>>>(col, dinv, E_EDGES);
    rsqrt_kernel<<<(N_NODES + TPB - 1) / TPB, TPB, 0, stream>>>(dinv, N_NODES);

    // --- conv1: xw = x @ W1 ; scatter ; relu(+b1) ---
    wmma_gemm_f32<<<dim3(1, N_NODES / 16, 1), 32, 0, stream>>>(x, W1, nullptr, bufA, S_IN, H_DIM, 0, S_IN);
    fill_kernel<<<(nh + TPB - 1) / TPB, TPB, 0, stream>>>(bufB, 0.0f, nh);
    scatter_kernel<<<(e16 + TPB - 1) / TPB, TPB, 0, stream>>>(bufA, row, col, dinv, bufB, e16);
    finalize_kernel<<<(nh + TPB - 1) / TPB, TPB, 0, stream>>>(bufB, bufA, dinv, b1, nh);

    // --- conv2 ---
    wmma_gemm_f32<<<dim3(1, N_NODES / 16, 1), 32, 0, stream>>>(bufB, W2, nullptr, bufA, H_DIM, H_DIM, 0, H_DIM);
    fill_kernel<<<(nh + TPB - 1) / TPB, TPB, 0, stream>>>(bufB, 0.0f, nh);
    scatter_kernel<<<(e16 + TPB - 1) / TPB, TPB, 0, stream>>>(bufA, row, col, dinv, bufB, e16);
    finalize_kernel<<<(nh + TPB - 1) / TPB, TPB, 0, stream>>>(bufB, bufA, dinv, b2, nh);

    // --- conv3 ---
    wmma_gemm_f32<<<dim3(1, N_NODES / 16, 1), 32, 0, stream>>>(bufB, W3, nullptr, bufA, H_DIM, H_DIM, 0, H_DIM);
    fill_kernel<<<(nh + TPB - 1) / TPB, TPB, 0, stream>>>(bufB, 0.0f, nh);
    scatter_kernel<<<(e16 + TPB - 1) / TPB, TPB, 0, stream>>>(bufA, row, col, dinv, bufB, e16);
    finalize_kernel<<<(nh + TPB - 1) / TPB, TPB, 0, stream>>>(bufB, bufA, dinv, b3, nh);

    // bufB is now h, viewed as [128, 16384] (row-major [N,16] flattens exactly)

    // --- FC trunk (split-K x8 for occupancy, deterministic reduction): ---
    // Hp = relu(h@Wp1+bp1)
    wmma_gemm_f32<<<dim3(256 / 16, B_GRAPHS / 16, KCHUNKS), 32, 0, stream>>>(
        bufB, Wp1, nullptr, part, FC_IN_SZ, 256, 0, FC_IN_SZ / KCHUNKS);
    reduce_bias_relu_kernel<<<(fcMN + TPB - 1) / TPB, TPB, 0, stream>>>(part, bp1, Hp, fcMN, 256, 1);
    // V = relu(h@Wv1+bv1)
    wmma_gemm_f32<<<dim3(256 / 16, B_GRAPHS / 16, KCHUNKS), 32, 0, stream>>>(
        bufB, Wv1, nullptr, part, FC_IN_SZ, 256, 0, FC_IN_SZ / KCHUNKS);
    reduce_bias_relu_kernel<<<(fcMN + TPB - 1) / TPB, TPB, 0, stream>>>(part, bv1, Vv, fcMN, 256, 1);

    // --- policy head: P = Hp @ Wp2 + bp2 ---
    wmma_gemm_f32<<<dim3(1024 / 16, B_GRAPHS / 16, 1), 32, 0, stream>>>(Hp, Wp2, bp2, P, 256, 1024, 0, 256);

    // --- value heads ---
    heads_kernel<<<1, B_GRAPHS, 0, stream>>>(Vv, Wiv, biv, Wev, bev, ext_V, int_V);
}